// ResidualTransformerLayer_41386304864864
// MI455X (gfx1250) — compile-verified
//
#include <hip/hip_runtime.h>
#include <math.h>

#define N_NODES 50000
#define N_EDGES 640000
#define HID     64
#define HEADS   4
#define DMODEL  256   // HID*HEADS

typedef __attribute__((ext_vector_type(2))) float v2f;
typedef __attribute__((ext_vector_type(8))) float v8f;

// ---- monotonic float <-> uint encoding for atomicMax on floats ----
__device__ __forceinline__ unsigned encodeF(float f) {
    int i = __float_as_int(f);
    return (i < 0) ? ~((unsigned)i) : ((unsigned)i | 0x80000000u);
}
__device__ __forceinline__ float decodeF(unsigned k) {
    return (k & 0x80000000u) ? __int_as_float((int)(k & 0x7FFFFFFFu))
                             : __int_as_float((int)(~k));
}

// ============================================================================
// Kernel 1: q/k/v/skip projections via V_WMMA_F32_16X16X4_F32.
// One wave per (row-tile, 16-col slab); computes the tile for ALL FOUR weight
// matrices with a single A-fragment load per K-step (4 B loads + 4 WMMAs).
// Tiles: 3125 row-tiles x 16 col-slabs -> 50000 waves, 64 WMMAs each.
// ============================================================================
__global__ void proj_wmma_kernel(const float* __restrict__ x,
                                 const float* __restrict__ Wq, const float* __restrict__ bq,
                                 const float* __restrict__ Wk, const float* __restrict__ bk,
                                 const float* __restrict__ Wv, const float* __restrict__ bv,
                                 const float* __restrict__ Ws, const float* __restrict__ bs,
                                 float* __restrict__ qo, float* __restrict__ ko_,
                                 float* __restrict__ vo, float* __restrict__ xro) {
    int wave = (blockIdx.x * blockDim.x + threadIdx.x) >> 5;
    int ct = wave & 15;          // 16-column slab within DMODEL
    int rt = wave >> 4;          // 16-node row tile
    if (rt >= N_NODES / 16) return;

    int lane = threadIdx.x & 31;
    int half = lane >> 4;        // 0: K pair {0,1}; 1: K pair {2,3}
    int l16  = lane & 15;
    int colb = ct << 4;
    int rb   = rt * 16;

    const float* arow = x  + (size_t)(rb   + l16) * HID;   // A: row = node, K along HID
    const float* bq_r = Wq + (size_t)(colb + l16) * HID;   // B(K,N) = W[N][K]
    const float* bk_r = Wk + (size_t)(colb + l16) * HID;
    const float* bv_r = Wv + (size_t)(colb + l16) * HID;
    const float* bs_r = Ws + (size_t)(colb + l16) * HID;

    v8f accq = {0.f,0.f,0.f,0.f,0.f,0.f,0.f,0.f};
    v8f acck = accq, accv = accq, accs = accq;

#pragma unroll
    for (int kk = 0; kk < HID; kk += 4) {
        int ka = kk + half * 2;
        v2f a, b0, b1, b2, b3;
        a.x  = arow[ka]; a.y  = arow[ka + 1];
        b0.x = bq_r[ka]; b0.y = bq_r[ka + 1];
        b1.x = bk_r[ka]; b1.y = bk_r[ka + 1];
        b2.x = bv_r[ka]; b2.y = bv_r[ka + 1];
        b3.x = bs_r[ka]; b3.y = bs_r[ka + 1];
        accq = __builtin_amdgcn_wmma_f32_16x16x4_f32(false, a, false, b0, (short)0, accq, false, false);
        acck = __builtin_amdgcn_wmma_f32_16x16x4_f32(false, a, false, b1, (short)0, acck, false, false);
        accv = __builtin_amdgcn_wmma_f32_16x16x4_f32(false, a, false, b2, (short)0, accv, false, false);
        accs = __builtin_amdgcn_wmma_f32_16x16x4_f32(false, a, false, b3, (short)0, accs, false, false);
    }

    int col = colb + l16;
    float cq = bq[col], ck = bk[col], cv = bv[col], cs = bs[col];
#pragma unroll
    for (int r = 0; r < 8; ++r) {
        size_t row = (size_t)(rb + half * 8 + r);  // C: lanes 0-15 -> M=r, 16-31 -> M=8+r
        qo [row * DMODEL + col] = accq[r] + cq;
        ko_[row * DMODEL + col] = acck[r] + ck;
        vo [row * DMODEL + col] = accv[r] + cv;
        xro[row * DMODEL + col] = accs[r] + cs;
    }
}

// ============================================================================
// Kernel 2: edge attention logits + per-dst max (encoded atomicMax).
// One wave per edge; lane covers 8 channels; 8 lanes per head.
// ============================================================================
__global__ void edge_logits_kernel(const int* __restrict__ ei,
                                   const float* __restrict__ q, const float* __restrict__ k,
                                   float* __restrict__ alpha, unsigned* __restrict__ mEnc) {
    int wave = (blockIdx.x * blockDim.x + threadIdx.x) >> 5;
    if (wave >= N_EDGES) return;
    int lane = threadIdx.x & 31;
    int src = ei[wave];
    int dst = ei[N_EDGES + wave];

    const float4* qp = reinterpret_cast<const float4*>(q + (size_t)dst * DMODEL + lane * 8);
    const float4* kp = reinterpret_cast<const float4*>(k + (size_t)src * DMODEL + lane * 8);
    float4 q0 = qp[0], q1 = qp[1];
    float4 k0 = kp[0], k1 = kp[1];
    float acc = q0.x*k0.x + q0.y*k0.y + q0.z*k0.z + q0.w*k0.w
              + q1.x*k1.x + q1.y*k1.y + q1.z*k1.z + q1.w*k1.w;
    acc += __shfl_xor(acc, 1, 32);
    acc += __shfl_xor(acc, 2, 32);
    acc += __shfl_xor(acc, 4, 32);
    if ((lane & 7) == 0) {
        int h = lane >> 3;
        float al = acc * 0.125f;           // 1/sqrt(HID)
        alpha[(size_t)wave * HEADS + h] = al;
        atomicMax(&mEnc[(size_t)dst * HEADS + h], encodeF(al));
    }
}

// ============================================================================
// Kernel 3: ex = exp(alpha - m[dst]) in place; atomicAdd into per-dst sum.
// ============================================================================
__global__ void edge_exp_kernel(const int* __restrict__ ei,
                                float* __restrict__ alpha,
                                const unsigned* __restrict__ mEnc,
                                float* __restrict__ ssum) {
    int i = blockIdx.x * blockDim.x + threadIdx.x;
    if (i >= N_EDGES * HEADS) return;
    int e = i >> 2;
    int h = i & 3;
    int dst = ei[N_EDGES + e];
    float mm = decodeF(mEnc[(size_t)dst * HEADS + h]);
    float ex = __expf(alpha[i] - mm);
    alpha[i] = ex;
    atomicAdd(&ssum[(size_t)dst * HEADS + h], ex);
}

// ============================================================================
// Kernel 4: weighted scatter-aggregate of v[src] into agg[dst].
// One wave per edge; 8 float atomics per lane.
// ============================================================================
__global__ void scatter_kernel(const int* __restrict__ ei,
                               const float* __restrict__ alpha,
                               const float* __restrict__ ssum,
                               const float* __restrict__ v,
                               float* __restrict__ agg) {
    int wave = (blockIdx.x * blockDim.x + threadIdx.x) >> 5;
    if (wave >= N_EDGES) return;
    int lane = threadIdx.x & 31;
    int src = ei[wave];
    int dst = ei[N_EDGES + wave];
    int h = lane >> 3;
    float attn = alpha[(size_t)wave * HEADS + h] /
                 (ssum[(size_t)dst * HEADS + h] + 1e-16f);
    const float4* vp = reinterpret_cast<const float4*>(v + (size_t)src * DMODEL + lane * 8);
    float4 v0 = vp[0], v1 = vp[1];
    float* op = agg + (size_t)dst * DMODEL + lane * 8;
    atomicAdd(op + 0, v0.x * attn);
    atomicAdd(op + 1, v0.y * attn);
    atomicAdd(op + 2, v0.z * attn);
    atomicAdd(op + 3, v0.w * attn);
    atomicAdd(op + 4, v1.x * attn);
    atomicAdd(op + 5, v1.y * attn);
    atomicAdd(op + 6, v1.z * attn);
    atomicAdd(op + 7, v1.w * attn);
}

// ============================================================================
// Kernel 5: beta-gated skip + LayerNorm.  One wave per node, 8 channels/lane.
// ============================================================================
__global__ void fuse_ln_kernel(const float* __restrict__ agg,
                               const float* __restrict__ xr,
                               const float* __restrict__ Wbeta,
                               const float* __restrict__ g,
                               const float* __restrict__ b,
                               float* __restrict__ hn) {
    int node = (blockIdx.x * blockDim.x + threadIdx.x) >> 5;
    if (node >= N_NODES) return;
    int lane = threadIdx.x & 31;
    int base = lane * 8;

    const float4* op4 = reinterpret_cast<const float4*>(agg + (size_t)node * DMODEL + base);
    const float4* xp4 = reinterpret_cast<const float4*>(xr  + (size_t)node * DMODEL + base);
    float4 oa = op4[0], ob = op4[1];
    float4 xa = xp4[0], xb = xp4[1];
    float o[8] = {oa.x, oa.y, oa.z, oa.w, ob.x, ob.y, ob.z, ob.w};
    float r[8] = {xa.x, xa.y, xa.z, xa.w, xb.x, xb.y, xb.z, xb.w};

    float dotb = 0.f;
#pragma unroll
    for (int j = 0; j < 8; ++j) {
        dotb += o[j] * Wbeta[base + j]
              + r[j] * Wbeta[DMODEL + base + j]
              + (o[j] - r[j]) * Wbeta[2 * DMODEL + base + j];
    }
    dotb += __shfl_xor(dotb, 1, 32);
    dotb += __shfl_xor(dotb, 2, 32);
    dotb += __shfl_xor(dotb, 4, 32);
    dotb += __shfl_xor(dotb, 8, 32);
    dotb += __shfl_xor(dotb, 16, 32);
    float beta = 1.f / (1.f + __expf(-dotb));

    float h[8];
    float s1 = 0.f, s2 = 0.f;
#pragma unroll
    for (int j = 0; j < 8; ++j) {
        h[j] = beta * r[j] + (1.f - beta) * o[j];
        s1 += h[j];
        s2 += h[j] * h[j];
    }
    s1 += __shfl_xor(s1, 1, 32);  s2 += __shfl_xor(s2, 1, 32);
    s1 += __shfl_xor(s1, 2, 32);  s2 += __shfl_xor(s2, 2, 32);
    s1 += __shfl_xor(s1, 4, 32);  s2 += __shfl_xor(s2, 4, 32);
    s1 += __shfl_xor(s1, 8, 32);  s2 += __shfl_xor(s2, 8, 32);
    s1 += __shfl_xor(s1, 16, 32); s2 += __shfl_xor(s2, 16, 32);

    float mu  = s1 * (1.f / DMODEL);
    float var = s2 * (1.f / DMODEL) - mu * mu;
    float rs  = rsqrtf(var + 1e-5f);

    float* hp = hn + (size_t)node * DMODEL + base;
#pragma unroll
    for (int j = 0; j < 8; ++j) {
        hp[j] = (h[j] - mu) * rs * g[base + j] + b[base + j];
    }
}

// ============================================================================
// Kernel 6: final projection hn[N,256] x Wproj^T[256,64] + bproj + x, ReLU.
// One wave per 16-node row tile computes ALL 64 output columns (4 col tiles):
// single A load per K-step, 4 B loads + 4 WMMAs. K=256 -> 256 WMMAs/wave.
// ============================================================================
__global__ void final_proj_wmma_kernel(const float* __restrict__ hn,
                                       const float* __restrict__ Wp,
                                       const float* __restrict__ bp,
                                       const float* __restrict__ x,
                                       float* __restrict__ out) {
    int rt = (blockIdx.x * blockDim.x + threadIdx.x) >> 5;
    if (rt >= N_NODES / 16) return;

    int lane = threadIdx.x & 31;
    int half = lane >> 4;
    int l16  = lane & 15;
    int rb   = rt * 16;

    const float* arow = hn + (size_t)(rb + l16) * DMODEL;
    const float* b0r  = Wp + (size_t)( 0 + l16) * DMODEL;   // Wproj is [64,256]
    const float* b1r  = Wp + (size_t)(16 + l16) * DMODEL;
    const float* b2r  = Wp + (size_t)(32 + l16) * DMODEL;
    const float* b3r  = Wp + (size_t)(48 + l16) * DMODEL;

    v8f acc0 = {0.f,0.f,0.f,0.f,0.f,0.f,0.f,0.f};
    v8f acc1 = acc0, acc2 = acc0, acc3 = acc0;

#pragma unroll 4
    for (int kk = 0; kk < DMODEL; kk += 4) {
        int ka = kk + half * 2;
        v2f a, b0, b1, b2, b3;
        a.x  = arow[ka]; a.y  = arow[ka + 1];
        b0.x = b0r[ka];  b0.y = b0r[ka + 1];
        b1.x = b1r[ka];  b1.y = b1r[ka + 1];
        b2.x = b2r[ka];  b2.y = b2r[ka + 1];
        b3.x = b3r[ka];  b3.y = b3r[ka + 1];
        acc0 = __builtin_amdgcn_wmma_f32_16x16x4_f32(false, a, false, b0, (short)0, acc0, false, false);
        acc1 = __builtin_amdgcn_wmma_f32_16x16x4_f32(false, a, false, b1, (short)0, acc1, false, false);
        acc2 = __builtin_amdgcn_wmma_f32_16x16x4_f32(false, a, false, b2, (short)0, acc2, false, false);
        acc3 = __builtin_amdgcn_wmma_f32_16x16x4_f32(false, a, false, b3, (short)0, acc3, false, false);
    }

    float c0 = bp[ 0 + l16], c1 = bp[16 + l16], c2 = bp[32 + l16], c3 = bp[48 + l16];
#pragma unroll
    for (int r = 0; r < 8; ++r) {
        size_t row = (size_t)(rb + half * 8 + r);
        const float* xrow = x   + row * HID;
        float*       orow = out + row * HID;
        orow[ 0 + l16] = fmaxf(acc0[r] + c0 + xrow[ 0 + l16], 0.f);
        orow[16 + l16] = fmaxf(acc1[r] + c1 + xrow[16 + l16], 0.f);
        orow[32 + l16] = fmaxf(acc2[r] + c2 + xrow[32 + l16], 0.f);
        orow[48 + l16] = fmaxf(acc3[r] + c3 + xrow[48 + l16], 0.f);
    }
}

// ============================================================================
extern "C" void kernel_launch(void* const* d_in, const int* in_sizes, int n_in,
                              void* d_out, int out_size, void* d_ws, size_t ws_size,
                              hipStream_t stream) {
    const float* x     = (const float*)d_in[0];
    const int*   ei    = (const int*)  d_in[1];
    const float* Wq    = (const float*)d_in[2];
    const float* bq    = (const float*)d_in[3];
    const float* Wk    = (const float*)d_in[4];
    const float* bk    = (const float*)d_in[5];
    const float* Wv    = (const float*)d_in[6];
    const float* bv    = (const float*)d_in[7];
    const float* Ws    = (const float*)d_in[8];
    const float* bs    = (const float*)d_in[9];
    const float* Wbeta = (const float*)d_in[10];
    const float* ln_g  = (const float*)d_in[11];
    const float* ln_b  = (const float*)d_in[12];
    const float* Wp    = (const float*)d_in[13];
    const float* bp    = (const float*)d_in[14];
    float* out = (float*)d_out;

    const size_t nD = (size_t)N_NODES * DMODEL;
    float* ws = (float*)d_ws;
    float*    q     = ws;                 // reused as hn after edge logits
    float*    k     = q + nD;
    float*    v     = k + nD;
    float*    xr    = v + nD;
    float*    agg   = xr + nD;
    float*    alpha = agg + nD;                                // E*HEADS
    float*    ssum  = alpha + (size_t)N_EDGES * HEADS;         // N*HEADS
    unsigned* mEnc  = (unsigned*)(ssum + (size_t)N_NODES * HEADS); // N*HEADS
    float*    hn    = q;   // alias: q is dead after edge_logits

    // zero per-call accumulators (graph-capture-safe)
    hipMemsetAsync(agg,  0, nD * sizeof(float), stream);
    hipMemsetAsync(ssum, 0, (size_t)N_NODES * HEADS * sizeof(float), stream);
    hipMemsetAsync(mEnc, 0, (size_t)N_NODES * HEADS * sizeof(unsigned), stream);

    // 1. projections: 3125 row-tiles x 16 col-slabs = 50000 waves, 8 waves/block
    {
        int waves = (N_NODES / 16) * 16;
        proj_wmma_kernel<<<waves / 8, 256, 0, stream>>>(
            x, Wq, bq, Wk, bk, Wv, bv, Ws, bs, q, k, v, xr);
    }
    // 2. edge logits + segment max
    edge_logits_kernel<<<N_EDGES / 8, 256, 0, stream>>>(ei, q, k, alpha, mEnc);
    // 3. exp + segment sum
    edge_exp_kernel<<<(N_EDGES * HEADS) / 256, 256, 0, stream>>>(ei, alpha, mEnc, ssum);
    // 4. weighted scatter-aggregate
    scatter_kernel<<<N_EDGES / 8, 256, 0, stream>>>(ei, alpha, ssum, v, agg);
    // 5. beta gate + LayerNorm
    fuse_ln_kernel<<<(N_NODES + 7) / 8, 256, 0, stream>>>(agg, xr, Wbeta, ln_g, ln_b, hn);
    // 6. final projection + residual + ReLU: one wave per row tile
    {
        int waves = N_NODES / 16;  // 3125
        final_proj_wmma_kernel<<<(waves + 7) / 8, 256, 0, stream>>>(hn, Wp, bp, x, out);
    }
}